// PartTripletLoss_80229989089611
// MI455X (gfx1250) — compile-verified
//
#include <hip/hip_runtime.h>
#include <math.h>

#define MARGIN 0.2f
#define NB 62      // batches (n)
#define M  256     // samples per batch (m)
#define D  256     // feature dim
#define PK 16      // positives per anchor (incl. self)

typedef __attribute__((ext_vector_type(2))) float v2f;
typedef __attribute__((ext_vector_type(8))) float v8f;

// ---------------- workspace layout (bytes) ----------------
// [0,            63488)      x2   : NB*M floats (squared row norms)
// [63488,        16316416)   dist : NB*M*M floats
// [16316416,     16316664)   bsum : NB floats
// [16316664,     16316912)   bcnt : NB floats
#define WS_X2    0
#define WS_DIST  63488
#define WS_BSUM  16316416
#define WS_BCNT  16316664

// -------- kernel 0: squared row norms x2[b][i] = sum_k X[b][i][k]^2 --------
__global__ void ptl_norms(const float* __restrict__ x, float* __restrict__ x2) {
    int b = blockIdx.x;          // 0..61
    int i = threadIdx.x;         // 0..255
    const float4* row = (const float4*)(x + ((size_t)b * M + i) * D);
    float s = 0.f;
    #pragma unroll 4
    for (int k = 0; k < D / 4; ++k) {
        float4 v = row[k];
        s += v.x * v.x + v.y * v.y + v.z * v.z + v.w * v.w;
    }
    x2[b * M + i] = s;
}

// -------- kernel 1: WMMA fp32 Gram + fused distance --------
// One wave (32 threads) per 16x16 tile of dist[b].
// grid = (16, 16, 62), block = 32. EXEC all-1s as WMMA requires.
__global__ void ptl_dist_wmma(const float* __restrict__ x,
                              const float* __restrict__ x2,
                              float* __restrict__ dist) {
    int ti = blockIdx.x;         // row tile
    int tj = blockIdx.y;         // col tile
    int b  = blockIdx.z;
    int lane = threadIdx.x;      // 0..31
    int r    = lane & 15;
    int half = lane >> 4;        // 0: K+{0,1}, 1: K+{2,3}

    const float* Xb = x + (size_t)b * M * D;
    // A fragment: rows ti*16 + r, cols k + 2*half + {0,1}
    const float* arow = Xb + (size_t)(ti * 16 + r) * D + 2 * half;
    // B fragment: B[k][n] = X[tj*16 + n][k] -> same per-lane pattern, j-tile base
    const float* brow = Xb + (size_t)(tj * 16 + r) * D + 2 * half;

    v8f acc = {0.f, 0.f, 0.f, 0.f, 0.f, 0.f, 0.f, 0.f};
    #pragma unroll 8
    for (int k = 0; k < D; k += 4) {
        v2f a  = *(const v2f*)(arow + k);
        v2f bb = *(const v2f*)(brow + k);
        // D = A x B + C ; emits v_wmma_f32_16x16x4_f32
        acc = __builtin_amdgcn_wmma_f32_16x16x4_f32(
            /*neg_a=*/false, a, /*neg_b=*/false, bb,
            /*c_mod=*/(short)0, acc, /*reuse_a=*/false, /*reuse_b=*/false);
    }

    // C/D layout: acc[v] holds row M = v + 8*half, col N = r (per lane).
    const float* x2b = x2 + b * M;
    float xj = x2b[tj * 16 + r];
    float* drow = dist + ((size_t)b * M + ti * 16 + half * 8) * M + tj * 16 + r;
    #pragma unroll
    for (int v = 0; v < 8; ++v) {
        float xi = x2b[ti * 16 + half * 8 + v];
        float sq = xi + xj - 2.f * acc[v];
        sq = sq > 0.f ? sq : 0.f;
        drow[(size_t)v * M] = sqrtf(sq);
    }
}

// -------- kernel 2: triplet pair loss; one wave per anchor --------
// grid = (32, 62), block = 256 (8 waves). Accumulates into bsum/bcnt[b].
__global__ void ptl_loss(const float* __restrict__ dist,
                         const int* __restrict__ label,
                         float* __restrict__ bsum,
                         float* __restrict__ bcnt) {
    int b    = blockIdx.y;
    int wave = threadIdx.x >> 5;
    int lane = threadIdx.x & 31;
    int anchor = blockIdx.x * 8 + wave;

    __shared__ float pos_d[8][PK];
    __shared__ int   pos_n[8];
    __shared__ float s_sum, s_cnt;
    if (threadIdx.x == 0) { s_sum = 0.f; s_cnt = 0.f; }
    if (lane == 0) pos_n[wave] = 0;
    __syncthreads();

    const int* lab = label + b * M;
    int myl = lab[anchor];
    const float* drow = dist + ((size_t)b * M + anchor) * M;

    float dj[8];
    int   neg[8];
    #pragma unroll
    for (int t = 0; t < 8; ++t) {
        int j = lane + 32 * t;
        float dv = drow[j];
        int lj = lab[j];
        dj[t]  = dv;
        neg[t] = (lj != myl);
        if (lj == myl) {                       // positive (incl. self, d=0)
            int idx = atomicAdd(&pos_n[wave], 1);
            if (idx < PK) pos_d[wave][idx] = dv;
        }
    }
    __syncthreads();

    int P = pos_n[wave]; if (P > PK) P = PK;
    float lsum = 0.f, lcnt = 0.f;
    #pragma unroll
    for (int t = 0; t < 8; ++t) {
        if (neg[t]) {
            float dn = dj[t];
            for (int p = 0; p < P; ++p) {
                float v = MARGIN + pos_d[wave][p] - dn;
                if (v > 0.f) { lsum += v; lcnt += 1.f; }
            }
        }
    }
    // wave32 reduction
    for (int off = 16; off; off >>= 1) {
        lsum += __shfl_down(lsum, off, 32);
        lcnt += __shfl_down(lcnt, off, 32);
    }
    if (lane == 0) { atomicAdd(&s_sum, lsum); atomicAdd(&s_cnt, lcnt); }
    __syncthreads();
    if (threadIdx.x == 0) {
        atomicAdd(&bsum[b], s_sum);
        atomicAdd(&bcnt[b], s_cnt);
    }
}

// -------- small helpers --------
__global__ void ptl_zero(float* p, int n) {
    int i = blockIdx.x * blockDim.x + threadIdx.x;
    if (i < n) p[i] = 0.f;
}

__global__ void ptl_final(const float* __restrict__ bsum,
                          const float* __restrict__ bcnt,
                          float* __restrict__ out) {
    if (threadIdx.x == 0) {
        float accL = 0.f, accC = 0.f;
        for (int b = 0; b < NB; ++b) {
            float c  = bcnt[b];
            float lm = (c > 0.f) ? (bsum[b] / c) : 0.f;
            accL += lm;
            accC += c;
        }
        out[0] = accL / (float)NB;
        out[1] = accC / (float)NB;
    }
}

extern "C" void kernel_launch(void* const* d_in, const int* in_sizes, int n_in,
                              void* d_out, int out_size, void* d_ws, size_t ws_size,
                              hipStream_t stream) {
    const float* feature = (const float*)d_in[0];   // (62, 256, 256) f32
    const int*   label   = (const int*)d_in[1];     // (62, 256) i32
    float* out = (float*)d_out;                     // [loss_mean, nonzero_mean]

    char* ws = (char*)d_ws;
    float* x2   = (float*)(ws + WS_X2);
    float* dist = (float*)(ws + WS_DIST);
    float* bsum = (float*)(ws + WS_BSUM);
    float* bcnt = (float*)(ws + WS_BCNT);

    // 0) squared row norms
    ptl_norms<<<dim3(NB), dim3(M), 0, stream>>>(feature, x2);

    // zero per-batch accumulators
    ptl_zero<<<dim3(1), dim3(128), 0, stream>>>(bsum, NB);
    ptl_zero<<<dim3(1), dim3(128), 0, stream>>>(bcnt, NB);

    // 1) WMMA fp32 Gram + fused sqrt(relu(...)) -> dist
    ptl_dist_wmma<<<dim3(M / 16, M / 16, NB), dim3(32), 0, stream>>>(feature, x2, dist);

    // 2) pair loss accumulation per batch
    ptl_loss<<<dim3(M / 8, NB), dim3(256), 0, stream>>>(dist, label, bsum, bcnt);

    // 3) finalize two scalars
    ptl_final<<<dim3(1), dim3(32), 0, stream>>>(bsum, bcnt, out);
}